// SimpleDynamicModel_7370163880075
// MI455X (gfx1250) — compile-verified
//
#include <hip/hip_runtime.h>

// Ensemble dynamics model (E=16, B=16384, I=96, HS=64, S=64, HR=32, R=1), fp32.
// Memory floor: ~275 MB HBM traffic -> ~12us @ 23.3 TB/s. 7 GFLOP total.
// Uses V_WMMA_F32_16X16X4_F32 (full fp32 matrix path) to match the fp32 reference.
// Outputs stored non-temporally so x + weights stay resident in the 192 MB L2.

typedef __attribute__((ext_vector_type(2))) float v2f;
typedef __attribute__((ext_vector_type(4))) float v4f;
typedef __attribute__((ext_vector_type(8))) float v8f;

constexpr int kE  = 16;
constexpr int kS  = 64;
constexpr int kA  = 32;
constexpr int kI  = 96;   // kS + kA
constexpr int kHS = 64;
constexpr int kHR = 32;
constexpr int kB  = 16384;
constexpr int kBT = 64;   // batch tile rows per workgroup (4 waves x 16)

// LDS strides (floats). Rows 16B aligned; 16-row column reads conflict-free
// on 64 banks:
//   xs: 100 -> row*36 mod 64 is a permutation over 16 rows
//   hs:  68 -> row*4  mod 64, stores and A-frag reads fully conflict-free
//   hr:  36 -> row*36 mod 64 permutation
constexpr int kXSS = 100;
constexpr int kHSS = 68;
constexpr int kHRS = 36;

static __device__ __forceinline__ v8f splat8(float v) {
  v8f x = {v, v, v, v, v, v, v, v};
  return x;
}

__global__ __launch_bounds__(128)
void ensemble_dyn_kernel(const float* __restrict__ state,
                         const float* __restrict__ action,
                         const float* __restrict__ W1,  const float* __restrict__ b1,
                         const float* __restrict__ W2,  const float* __restrict__ b2,
                         const float* __restrict__ rW1, const float* __restrict__ rb1,
                         const float* __restrict__ rW2, const float* __restrict__ rb2,
                         float* __restrict__ pred_states,
                         float* __restrict__ pred_rewards) {
  __shared__ __align__(16) float xs[kBT * kXSS];  // x tile  [64 x 96]
  __shared__ __align__(16) float hs[kBT * kHSS];  // h tile  [64 x 64]
  __shared__ __align__(16) float hr[kBT * kHRS];  // hr tile [64 x 32]

  const int e    = blockIdx.y;
  const int row0 = blockIdx.x * kBT;
  const int tid  = threadIdx.x;

  // ---- cooperative, coalesced load of x = [state | action] tile into LDS ----
  for (int i = tid; i < kBT * 16; i += 128) {        // state: 64 rows x 16 float4
    int r = i >> 4, q = i & 15;
    v4f v = *(const v4f*)(state + (size_t)(row0 + r) * kS + q * 4);
    *(v4f*)(&xs[r * kXSS + q * 4]) = v;
  }
  for (int i = tid; i < kBT * 8; i += 128) {         // action: 64 rows x 8 float4
    int r = i >> 3, q = i & 7;
    v4f v = *(const v4f*)(action + (size_t)(row0 + r) * kA + q * 4);
    *(v4f*)(&xs[r * kXSS + kS + q * 4]) = v;
  }
  __syncthreads();

  const int lane = tid & 31;   // wave32
  const int wave = tid >> 5;   // 0..3
  const int m0   = wave * 16;  // M-stripe within tile (private per wave)
  const int col  = lane & 15;  // N index inside a 16x16 tile (C/D/B layout)
  const int half = lane >> 4;  // 0/1: K-half selector (A/B layout)
  const int arow = m0 + col;   // A-fragment row (lane indexes M for A)

  // ======================= state model, layer 1: h = relu(x@W1 + b1) ========
  {
    const float* W1e = W1 + (size_t)e * kI * kHS;
    v8f acc[4];
#pragma unroll
    for (int j = 0; j < 4; ++j) acc[j] = splat8(b1[e * kHS + j * 16 + col]);

    for (int k0 = 0; k0 < kI; k0 += 4) {
      // A frag: reg0 = x[row][k0+2h], reg1 = x[row][k0+1+2h]  -> one ds_load_b64
      v2f a = *(const v2f*)(&xs[arow * kXSS + k0 + 2 * half]);
#pragma unroll
      for (int j = 0; j < 4; ++j) {
        v2f b;
        b.x = W1e[(size_t)(k0 + 2 * half) * kHS + j * 16 + col];
        b.y = W1e[(size_t)(k0 + 2 * half + 1) * kHS + j * 16 + col];
        acc[j] = __builtin_amdgcn_wmma_f32_16x16x4_f32(
            false, a, false, b, (short)0, acc[j], false, false);
      }
    }
    // ReLU + stash h in LDS (D layout: reg r -> row m0 + r + 8*half, col 16j+col)
#pragma unroll
    for (int j = 0; j < 4; ++j)
#pragma unroll
      for (int r = 0; r < 8; ++r)
        hs[(m0 + r + 8 * half) * kHSS + j * 16 + col] = fmaxf(acc[j][r], 0.f);
  }

  // ======================= state model, layer 2: out = h@W2 + b2 ============
  {
    const float* W2e = W2 + (size_t)e * kHS * kS;
    v8f acc[4];
#pragma unroll
    for (int j = 0; j < 4; ++j) acc[j] = splat8(b2[e * kS + j * 16 + col]);

    for (int k0 = 0; k0 < kHS; k0 += 4) {
      v2f a = *(const v2f*)(&hs[arow * kHSS + k0 + 2 * half]);
#pragma unroll
      for (int j = 0; j < 4; ++j) {
        v2f b;
        b.x = W2e[(size_t)(k0 + 2 * half) * kS + j * 16 + col];
        b.y = W2e[(size_t)(k0 + 2 * half + 1) * kS + j * 16 + col];
        acc[j] = __builtin_amdgcn_wmma_f32_16x16x4_f32(
            false, a, false, b, (short)0, acc[j], false, false);
      }
    }
    // store pred_states [E, B, S] non-temporally (pure streaming output)
    float* outp = pred_states + ((size_t)e * kB + row0) * kS;
#pragma unroll
    for (int j = 0; j < 4; ++j)
#pragma unroll
      for (int r = 0; r < 8; ++r)
        __builtin_nontemporal_store(
            acc[j][r],
            outp + (size_t)(m0 + r + 8 * half) * kS + j * 16 + col);
  }

  // ======================= reward model ======================================
  {
    const float* rW1e = rW1 + (size_t)e * kI * kHR;
    v8f acc[2];
#pragma unroll
    for (int j = 0; j < 2; ++j) acc[j] = splat8(rb1[e * kHR + j * 16 + col]);

    for (int k0 = 0; k0 < kI; k0 += 4) {
      v2f a = *(const v2f*)(&xs[arow * kXSS + k0 + 2 * half]);
#pragma unroll
      for (int j = 0; j < 2; ++j) {
        v2f b;
        b.x = rW1e[(size_t)(k0 + 2 * half) * kHR + j * 16 + col];
        b.y = rW1e[(size_t)(k0 + 2 * half + 1) * kHR + j * 16 + col];
        acc[j] = __builtin_amdgcn_wmma_f32_16x16x4_f32(
            false, a, false, b, (short)0, acc[j], false, false);
      }
    }
#pragma unroll
    for (int j = 0; j < 2; ++j)
#pragma unroll
      for (int r = 0; r < 8; ++r)
        hr[(m0 + r + 8 * half) * kHRS + j * 16 + col] = fmaxf(acc[j][r], 0.f);

    // Final 32->1 GEMV: lane pair (col, col+16) splits K; reduce via shfl_xor.
    const float* rW2e = rW2 + (size_t)e * kHR;  // [HR,1]
    float sum = 0.f;
#pragma unroll
    for (int k = 0; k < 16; ++k)
      sum += hr[(m0 + col) * kHRS + half * 16 + k] * rW2e[half * 16 + k];
    sum += __shfl_xor(sum, 16, 32);
    if (half == 0) {
      float outv = sum + rb2[e];
      outv = fminf(fmaxf(outv, -100.f), 200.f);
      __builtin_nontemporal_store(outv,
                                  pred_rewards + (size_t)e * kB + row0 + m0 + col);
    }
  }
}

extern "C" void kernel_launch(void* const* d_in, const int* in_sizes, int n_in,
                              void* d_out, int out_size, void* d_ws, size_t ws_size,
                              hipStream_t stream) {
  const float* state  = (const float*)d_in[0];
  const float* action = (const float*)d_in[1];
  const float* W1     = (const float*)d_in[2];
  const float* b1     = (const float*)d_in[3];
  const float* W2     = (const float*)d_in[4];
  const float* b2     = (const float*)d_in[5];
  const float* rW1    = (const float*)d_in[6];
  const float* rb1    = (const float*)d_in[7];
  const float* rW2    = (const float*)d_in[8];
  const float* rb2    = (const float*)d_in[9];

  float* pred_states  = (float*)d_out;                       // [E, B, S]
  float* pred_rewards = pred_states + (size_t)kE * kB * kS;  // [E, B, 1]

  dim3 grid(kB / kBT, kE);
  dim3 block(128);
  ensemble_dyn_kernel<<<grid, block, 0, stream>>>(
      state, action, W1, b1, W2, b2, rW1, rb1, rW2, rb2,
      pred_states, pred_rewards);
}